// EdgeClassificationGNN_17394617549531
// MI455X (gfx1250) — compile-verified
//
#include <hip/hip_runtime.h>
#include <hip/hip_bf16.h>
#include <math.h>

typedef __attribute__((ext_vector_type(2))) float v2f;
typedef __attribute__((ext_vector_type(8))) float v8f;

#define HID 128
#define HEADS 2
#define DH 256            // HEADS*HID
#define NEG_SLOPE 0.2f

// ---------------------------------------------------------------------------
// Node-feature GEMM: H[n, 0:256] = A[n, 0:K] @ W[K, 256]
// Block = 512 threads = 16 waves. Block computes rows [16*bid, 16*bid+16),
// wave w computes cols [16w, 16w+16) via V_WMMA_F32_16X16X4_F32 over K.
// CONCAT=true: A is concat(x[n,0:64], rep[n,0:64]) with K=128.
// ---------------------------------------------------------------------------
template<int K, bool CONCAT>
__global__ __launch_bounds__(512)
void gemm_node(const float* __restrict__ A, const float* __restrict__ A2,
               const float* __restrict__ W, float* __restrict__ Hout, int n)
{
    constexpr int PITCH = K + 4;           // pad to avoid 64-bank conflicts
    __shared__ float As[16 * PITCH];

    const int tid  = threadIdx.x;
    const int wid  = tid >> 5;
    const int lane = tid & 31;
    const int r0   = blockIdx.x * 16;

    // cooperative stage of 16 x K A-slab (coalesced along K)
    for (int idx = tid; idx < 16 * K; idx += 512) {
        const int row = idx / K;
        const int k   = idx - row * K;
        const int gr  = r0 + row;
        float v = 0.0f;
        if (gr < n) {
            if (CONCAT) v = (k < 64) ? A[gr * 64 + k] : A2[gr * 64 + (k - 64)];
            else        v = A[gr * K + k];
        }
        As[row * PITCH + k] = v;
    }
    __syncthreads();

    const int c0    = wid * 16;
    const int halfl = lane & 15;
    const int koff  = (lane >> 4) * 2;     // lanes 0-15: K+0/+1 ; lanes 16-31: K+2/+3

    v8f acc = {};
    for (int k0 = 0; k0 < K; k0 += 4) {
        v2f a, b;
        a.x = As[halfl * PITCH + k0 + koff];
        a.y = As[halfl * PITCH + k0 + koff + 1];
        b.x = W[(k0 + koff)     * DH + c0 + halfl];
        b.y = W[(k0 + koff + 1) * DH + c0 + halfl];
        acc = __builtin_amdgcn_wmma_f32_16x16x4_f32(
                  false, a, false, b, (short)0, acc, false, false);
    }

    // C/D layout: VGPR r -> M = r (lanes 0-15) or 8+r (lanes 16-31), N = lane&15
    const int colg  = c0 + halfl;
    const int mbase = (lane >> 4) * 8;
#pragma unroll
    for (int r = 0; r < 8; ++r) {
        const int grow = r0 + mbase + r;
        if (grow < n) Hout[grow * DH + colg] = acc[r];
    }
}

// ---------------------------------------------------------------------------
// Per-node attention scores: el[n,h] = <H[n,h,:], al[h,:]> ; er likewise.
// One wave per node, shuffle reduction.
// ---------------------------------------------------------------------------
__global__ __launch_bounds__(256)
void node_scores(const float* __restrict__ H, const float* __restrict__ al,
                 const float* __restrict__ ar, float* __restrict__ el,
                 float* __restrict__ er, int n)
{
    const int lane = threadIdx.x & 31;
    const int node = blockIdx.x * 8 + (threadIdx.x >> 5);
    if (node >= n) return;

    const float* h = H + node * DH;
    float e0 = 0.f, e1 = 0.f, r0 = 0.f, r1 = 0.f;
    for (int i = lane; i < HID; i += 32) {
        const float h0 = h[i], h1 = h[HID + i];
        e0 += h0 * al[i];        e1 += h1 * al[HID + i];
        r0 += h0 * ar[i];        r1 += h1 * ar[HID + i];
    }
#pragma unroll
    for (int off = 16; off; off >>= 1) {
        e0 += __shfl_xor(e0, off); e1 += __shfl_xor(e1, off);
        r0 += __shfl_xor(r0, off); r1 += __shfl_xor(r1, off);
    }
    if (lane == 0) {
        el[node * 2 + 0] = e0; el[node * 2 + 1] = e1;
        er[node * 2 + 0] = r0; er[node * 2 + 1] = r1;
    }
}

// ---------------------------------------------------------------------------
// Per-layer accumulator init (must run every launch; harness doesn't re-zero)
// ---------------------------------------------------------------------------
__global__ void init_layer(float* __restrict__ agg, float* __restrict__ m,
                           float* __restrict__ s, int n)
{
    const int i = blockIdx.x * 256 + threadIdx.x;
    if (i < n * DH) agg[i] = 0.0f;
    if (i < n * HEADS) { m[i] = -INFINITY; s[i] = 0.0f; }
}

__device__ __forceinline__ float lrelu(float x) {
    return x > 0.0f ? x : NEG_SLOPE * x;
}

// float atomic-max via int/uint ordering trick (init value is -inf)
__device__ __forceinline__ void atomicMaxF(float* addr, float val) {
    if (val >= 0.0f) atomicMax((int*)addr, __float_as_int(val));
    else             atomicMin((unsigned int*)addr, __float_as_uint(val));
}

__global__ void edge_max(const int* __restrict__ src, const int* __restrict__ dst,
                         const float* __restrict__ el, const float* __restrict__ er,
                         float* __restrict__ m, int e)
{
    const int i = blockIdx.x * 256 + threadIdx.x;
    if (i >= e) return;
    const int s = src[i], d = dst[i];
#pragma unroll
    for (int h = 0; h < HEADS; ++h)
        atomicMaxF(&m[d * 2 + h], lrelu(el[s * 2 + h] + er[d * 2 + h]));
}

__global__ void m_fixup(float* __restrict__ m, int n2)
{
    const int i = blockIdx.x * 256 + threadIdx.x;
    if (i < n2 && !isfinite(m[i])) m[i] = 0.0f;
}

__global__ void edge_exp(const int* __restrict__ src, const int* __restrict__ dst,
                         const float* __restrict__ el, const float* __restrict__ er,
                         const float* __restrict__ m, float* __restrict__ ex,
                         float* __restrict__ ssum, int e)
{
    const int i = blockIdx.x * 256 + threadIdx.x;
    if (i >= e) return;
    const int s = src[i], d = dst[i];
#pragma unroll
    for (int h = 0; h < HEADS; ++h) {
        const float sc = lrelu(el[s * 2 + h] + er[d * 2 + h]);
        const float v  = __expf(sc - m[d * 2 + h]);
        ex[i * 2 + h] = v;
        atomicAdd(&ssum[d * 2 + h], v);
    }
}

// wave per edge: agg[dst, j] += alpha(head(j)) * H[src, j],  j in [0,256)
__global__ __launch_bounds__(256)
void edge_agg(const int* __restrict__ src, const int* __restrict__ dst,
              const float* __restrict__ ex, const float* __restrict__ ssum,
              const float* __restrict__ H, float* __restrict__ agg, int e)
{
    const int w    = blockIdx.x * 8 + (threadIdx.x >> 5);
    const int lane = threadIdx.x & 31;
    if (w >= e) return;
    const int s = src[w], d = dst[w];
    const float w0 = ex[w * 2 + 0] / ssum[d * 2 + 0];
    const float w1 = ex[w * 2 + 1] / ssum[d * 2 + 1];
    const float* hs = H + s * DH;
    float* ad = agg + d * DH;
#pragma unroll
    for (int i = 0; i < 8; ++i) {
        const int j = i * 32 + lane;
        const float wt = (j < HID) ? w0 : w1;
        atomicAdd(&ad[j], wt * hs[j]);
    }
}

__global__ void bias_relu(const float* __restrict__ agg, const float* __restrict__ b,
                          float* __restrict__ outp, int n)
{
    const int i = blockIdx.x * 256 + threadIdx.x;
    if (i >= n * DH) return;
    const float v = agg[i] + b[i & (DH - 1)];
    outp[i] = v > 0.0f ? v : 0.0f;
}

// wave per edge: out[e,:2] = [H[src]||H[dst]] @ fcW + fcb
__global__ __launch_bounds__(256)
void edge_classify(const int* __restrict__ src, const int* __restrict__ dst,
                   const float* __restrict__ H, const float* __restrict__ fcW,
                   const float* __restrict__ fcb, float* __restrict__ out, int e)
{
    const int w    = blockIdx.x * 8 + (threadIdx.x >> 5);
    const int lane = threadIdx.x & 31;
    if (w >= e) return;
    const int s = src[w], d = dst[w];
    const float* hs = H + s * DH;
    const float* hd = H + d * DH;
    float a0 = 0.f, a1 = 0.f;
    for (int j = lane; j < DH; j += 32) {
        const float vs = hs[j], vd = hd[j];
        a0 += vs * fcW[j * 2 + 0] + vd * fcW[(DH + j) * 2 + 0];
        a1 += vs * fcW[j * 2 + 1] + vd * fcW[(DH + j) * 2 + 1];
    }
#pragma unroll
    for (int off = 16; off; off >>= 1) {
        a0 += __shfl_xor(a0, off);
        a1 += __shfl_xor(a1, off);
    }
    if (lane == 0) {
        out[w * 2 + 0] = a0 + fcb[0];
        out[w * 2 + 1] = a1 + fcb[1];
    }
}

// ---------------------------------------------------------------------------
extern "C" void kernel_launch(void* const* d_in, const int* in_sizes, int n_in,
                              void* d_out, int out_size, void* d_ws, size_t ws_size,
                              hipStream_t stream)
{
    const float* x   = (const float*)d_in[0];
    const float* rep = (const float*)d_in[1];
    const int*   src = (const int*)d_in[2];
    const int*   dst = (const int*)d_in[3];
    const float* W1  = (const float*)d_in[4];
    const float* al1 = (const float*)d_in[5];
    const float* ar1 = (const float*)d_in[6];
    const float* b1  = (const float*)d_in[7];
    const float* W2  = (const float*)d_in[8];
    const float* al2 = (const float*)d_in[9];
    const float* ar2 = (const float*)d_in[10];
    const float* b2  = (const float*)d_in[11];
    const float* fcW = (const float*)d_in[12];
    const float* fcb = (const float*)d_in[13];
    float* out = (float*)d_out;

    const int N = in_sizes[0] / 64;   // 25000
    const int E = in_sizes[2];        // 400000

    // workspace layout (floats)
    float* ws  = (float*)d_ws;
    float* h   = ws;                  // N*256  pre-aggregation features
    float* agg = h   + (size_t)N * DH;
    float* rly = agg + (size_t)N * DH; // relu(agg+bias)
    float* el  = rly + (size_t)N * DH;
    float* er  = el  + (size_t)N * 2;
    float* mbf = er  + (size_t)N * 2;
    float* sbf = mbf + (size_t)N * 2;
    float* ex  = sbf + (size_t)N * 2;  // E*2

    const int gN16  = (N + 15) / 16;
    const int gN8   = (N + 7) / 8;
    const int gE8   = (E + 7) / 8;
    const int gE256 = (E + 255) / 256;
    const int gF    = (N * DH + 255) / 256;
    const int gM    = (N * 2 + 255) / 256;

    // ---- layer 1 ----
    gemm_node<128, true><<<gN16, 512, 0, stream>>>(x, rep, W1, h, N);
    node_scores<<<gN8, 256, 0, stream>>>(h, al1, ar1, el, er, N);
    init_layer<<<gF, 256, 0, stream>>>(agg, mbf, sbf, N);
    edge_max<<<gE256, 256, 0, stream>>>(src, dst, el, er, mbf, E);
    m_fixup<<<gM, 256, 0, stream>>>(mbf, N * 2);
    edge_exp<<<gE256, 256, 0, stream>>>(src, dst, el, er, mbf, ex, sbf, E);
    edge_agg<<<gE8, 256, 0, stream>>>(src, dst, ex, sbf, h, agg, E);
    bias_relu<<<gF, 256, 0, stream>>>(agg, b1, rly, N);

    // ---- layer 2 ----
    gemm_node<256, false><<<gN16, 512, 0, stream>>>(rly, nullptr, W2, h, N);
    node_scores<<<gN8, 256, 0, stream>>>(h, al2, ar2, el, er, N);
    init_layer<<<gF, 256, 0, stream>>>(agg, mbf, sbf, N);
    edge_max<<<gE256, 256, 0, stream>>>(src, dst, el, er, mbf, E);
    m_fixup<<<gM, 256, 0, stream>>>(mbf, N * 2);
    edge_exp<<<gE256, 256, 0, stream>>>(src, dst, el, er, mbf, ex, sbf, E);
    edge_agg<<<gE8, 256, 0, stream>>>(src, dst, ex, sbf, h, agg, E);
    bias_relu<<<gF, 256, 0, stream>>>(agg, b2, rly, N);

    // ---- edge classifier ----
    edge_classify<<<gE8, 256, 0, stream>>>(src, dst, rly, fcW, fcb, out, E);
}